// PCLLoss_22969485099281
// MI455X (gfx1250) — compile-verified
//
#include <hip/hip_runtime.h>

// Problem constants (match reference): C=256 classes, N=64 samples, D=1024 dims.
#define C_CLASSES 256
#define N_SAMP    64
#define D_DIM     1024

typedef __attribute__((ext_vector_type(2))) float v2f;
typedef __attribute__((ext_vector_type(8))) float v8f;

// -----------------------------------------------------------------------------
// Kernel 1: per-class mean of L2-normalized rows.
// One block per (tensor, class): 512 blocks x 256 threads. Each thread owns 4
// consecutive floats (float4) of the D=1024 row -> fully coalesced 512B/wave
// loads. Norm uses an LDS tree reduction; the running mean lives in registers
// (no atomics -> deterministic). Bandwidth-bound: 134 MB total read.
// -----------------------------------------------------------------------------
__global__ __launch_bounds__(256) void class_means_kernel(
    const float* __restrict__ emb, const float* __restrict__ feat,
    float* __restrict__ mv, float* __restrict__ mt)
{
    __shared__ float red[256];
    const int b   = blockIdx.x;          // 0..511
    const int t   = b >> 8;              // 0 = embedding, 1 = features
    const int c   = b & (C_CLASSES - 1);
    const float* __restrict__ src = t ? feat : emb;
    float* __restrict__ dst       = t ? mt   : mv;
    const int tid = threadIdx.x;

    float ax = 0.f, ay = 0.f, az = 0.f, aw = 0.f;
    for (int n = 0; n < N_SAMP; ++n) {
        const float4* row = (const float4*)(src + ((size_t)c * N_SAMP + n) * D_DIM);
        float4 x = row[tid];
        red[tid] = x.x * x.x + x.y * x.y + x.z * x.z + x.w * x.w;
        __syncthreads();
        for (int s = 128; s > 0; s >>= 1) {
            if (tid < s) red[tid] += red[tid + s];
            __syncthreads();
        }
        const float nrm = fmaxf(sqrtf(red[0]), 1e-12f); // EPS_NORM on the norm
        __syncthreads();                                 // red[] reusable next iter
        const float inv = 1.0f / nrm;
        ax += x.x * inv; ay += x.y * inv; az += x.z * inv; aw += x.w * inv;
    }
    const float s = 1.0f / (float)N_SAMP;
    float4 r; r.x = ax * s; r.y = ay * s; r.z = az * s; r.w = aw * s;
    ((float4*)(dst + (size_t)c * D_DIM))[tid] = r;
}

// -----------------------------------------------------------------------------
// Kernel 2: symmetric Gram matrices pred = mv@mv^T, true = mt@mt^T via
// V_WMMA_F32_16X16X4_F32 (full fp32 — GEMM is only 0.27 GFLOP so precision
// beats fp16 throughput; keeps the downstream argsort order exact).
// One wave (32 threads) per 16x16 output tile: 512 blocks. EXEC is all-ones,
// no divergence around the WMMA (lane-half K-select is pure address math).
//
// A-matrix 16x4 f32 layout (ISA 7.12.2): lane L holds M = L&15; VGPR0/1 hold
// {K0,K1} for lanes 0-15 and {K2,K3} for lanes 16-31 -> per-lane contiguous
// float2 at column offset kh = (L>>4)*2. B = A^T of the J-tile rows, same
// per-lane addressing with the J row base (Gram symmetry).
// -----------------------------------------------------------------------------
__global__ __launch_bounds__(32) void gram_wmma_kernel(
    const float* __restrict__ mv, const float* __restrict__ mt,
    float* __restrict__ pred, float* __restrict__ tru)
{
    const int bid  = blockIdx.x;                 // 0..511
    const float* __restrict__ A = (bid >> 8) ? mt  : mv;
    float* __restrict__ O       = (bid >> 8) ? tru : pred;
    const int tile = bid & 255;
    const int I    = (tile >> 4) << 4;           // row-tile base
    const int J    = (tile & 15) << 4;           // col-tile base
    const int lane = threadIdx.x;                // 0..31
    const int m    = lane & 15;
    const int kh   = (lane >> 4) << 1;           // 0 or 2

    const float* arow = A + (size_t)(I + m) * D_DIM + kh;
    const float* brow = A + (size_t)(J + m) * D_DIM + kh;

    v8f acc = {};
    for (int k = 0; k < D_DIM; k += 4) {
        v2f a = *(const v2f*)(arow + k);
        v2f b = *(const v2f*)(brow + k);
        // 8 args: (neg_a, A, neg_b, B, c_mod, C, reuse_a, reuse_b)
        acc = __builtin_amdgcn_wmma_f32_16x16x4_f32(
            false, a, false, b, (short)0, acc, false, false);
    }

    // C/D layout: VGPR r -> row M = r (lanes 0-15) / r+8 (lanes 16-31), N = lane&15
    const int rbase = I + ((lane >> 4) << 3);
    const int col   = J + m;
#pragma unroll
    for (int r = 0; r < 8; ++r)
        O[(size_t)(rbase + r) * C_CLASSES + col] = acc[r];
}

// -----------------------------------------------------------------------------
// Kernel 3: ListMLE per row. One 256-thread block per row (256 blocks).
// Bitonic sort in LDS, descending by true-value with ascending-index tiebreak
// (matches jnp stable argsort deterministically), then exp -> inclusive suffix
// scan (Hillis-Steele) -> log(cs+eps)-ps -> block-reduce to rowloss[row].
// -----------------------------------------------------------------------------
__global__ __launch_bounds__(256) void listmle_kernel(
    const float* __restrict__ pred, const float* __restrict__ tru,
    float* __restrict__ rowloss)
{
    __shared__ float key[256];   // sort key (true), later reused as scan buffer
    __shared__ float val[256];   // payload (pred), later reused for reduction
    __shared__ int   idx[256];   // original index (stable tiebreak)
    const int row = blockIdx.x;
    const int i   = threadIdx.x;

    key[i] = tru [(size_t)row * C_CLASSES + i];
    val[i] = pred[(size_t)row * C_CLASSES + i];
    idx[i] = i;
    __syncthreads();

    // Bitonic sort: overall descending by key, tiebreak ascending idx.
    for (int k = 2; k <= 256; k <<= 1) {
        for (int j = k >> 1; j > 0; j >>= 1) {
            const int ixj = i ^ j;
            if (ixj > i) {
                const bool desc = ((i & k) == 0);     // this segment's direction
                const float ka = key[i], kb = key[ixj];
                const int   ia = idx[i], ib = idx[ixj];
                const bool inOrder = (ka > kb) || (ka == kb && ia < ib);
                if (inOrder != desc) {
                    key[i] = kb; key[ixj] = ka;
                    const float va = val[i]; val[i] = val[ixj]; val[ixj] = va;
                    idx[i] = ib; idx[ixj] = ia;
                }
            }
            __syncthreads();
        }
    }

    const float ps = val[i];     // pred sorted by descending true
    key[i] = expf(ps);
    __syncthreads();
    // Inclusive suffix sum: cs[i] = sum_{t>=i} exp(ps[t])
    for (int off = 1; off < 256; off <<= 1) {
        float v = key[i];
        if (i + off < 256) v += key[i + off];
        __syncthreads();
        key[i] = v;
        __syncthreads();
    }
    const float cs = key[i];
    val[i] = logf(cs + 1e-10f) - ps;   // EPS_LOG
    __syncthreads();
    for (int s = 128; s > 0; s >>= 1) {
        if (i < s) val[i] += val[i + s];
        __syncthreads();
    }
    if (i == 0) rowloss[row] = val[0];
}

// -----------------------------------------------------------------------------
// Kernel 4: final mean over 256 row losses -> d_out[0]. Separate kernel so we
// never rely on d_out being pre-zeroed (harness poisons it with 0xAA).
// -----------------------------------------------------------------------------
__global__ __launch_bounds__(256) void mean_kernel(
    const float* __restrict__ rowloss, float* __restrict__ out)
{
    __shared__ float red[256];
    const int i = threadIdx.x;
    red[i] = rowloss[i];
    __syncthreads();
    for (int s = 128; s > 0; s >>= 1) {
        if (i < s) red[i] += red[i + s];
        __syncthreads();
    }
    if (i == 0) out[0] = red[0] * (1.0f / (float)C_CLASSES);
}

extern "C" void kernel_launch(void* const* d_in, const int* in_sizes, int n_in,
                              void* d_out, int out_size, void* d_ws, size_t ws_size,
                              hipStream_t stream)
{
    const float* emb  = (const float*)d_in[0];   // [256,64,1024] f32
    const float* feat = (const float*)d_in[1];   // [256,64,1024] f32

    // Workspace layout (floats): mv | mt | pred | true | rowloss  (~2.6 MB)
    float* ws      = (float*)d_ws;
    float* mv      = ws;
    float* mt      = mv + (size_t)C_CLASSES * D_DIM;
    float* pred    = mt + (size_t)C_CLASSES * D_DIM;
    float* tru     = pred + (size_t)C_CLASSES * C_CLASSES;
    float* rowloss = tru  + (size_t)C_CLASSES * C_CLASSES;

    class_means_kernel<<<2 * C_CLASSES, 256, 0, stream>>>(emb, feat, mv, mt);
    gram_wmma_kernel  <<<2 * C_CLASSES, 32, 0, stream>>>(mv, mt, pred, tru);
    listmle_kernel    <<<C_CLASSES, 256, 0, stream>>>(pred, tru, rowloss);
    mean_kernel       <<<1, 256, 0, stream>>>(rowloss, (float*)d_out);
}